// GatedGraphConvNet_88553635709225
// MI455X (gfx1250) — compile-verified
//
#include <hip/hip_runtime.h>
#include <hip/hip_bf16.h>
#include <math.h>

typedef __attribute__((ext_vector_type(2))) float v2f;
typedef __attribute__((ext_vector_type(8))) float v8f;

#define N_NODES 100000
#define N_EDGES 1600000

// ---------------------------------------------------------------------------
__device__ __forceinline__ float elu1(float x) {
    return x > 0.0f ? x : (expf(x) - 1.0f);
}

// float atomic max via sign-split integer atomics (init must be -inf)
__device__ __forceinline__ void atomicMaxF(float* addr, float val) {
    if (val >= 0.0f) {
        atomicMax((int*)addr, __float_as_int(val));
    } else {
        atomicMin((unsigned int*)addr, __float_as_uint(val));
    }
}

// ---------------------------------------------------------------------------
// pad x (N x 16) -> h1 (N x 32), zeros in cols 16..31
__global__ void pad16to32_kernel(const float* __restrict__ x, float* __restrict__ h1, int n) {
    int i = blockIdx.x * blockDim.x + threadIdx.x;
    if (i >= n * 32) return;
    int node = i >> 5, j = i & 31;
    h1[i] = (j < 16) ? x[node * 16 + j] : 0.0f;
}

// h2 = [elu(h1), 0]  (N x 32 -> N x 64)
__global__ void elu_pad32to64_kernel(const float* __restrict__ h1, float* __restrict__ h2, int n) {
    int i = blockIdx.x * blockDim.x + threadIdx.x;
    if (i >= n * 64) return;
    int node = i >> 6, j = i & 63;
    h2[i] = (j < 32) ? elu1(h1[node * 32 + j]) : 0.0f;
}

__global__ void fill_kernel(float* __restrict__ p, float v, long n) {
    long i = (long)blockIdx.x * blockDim.x + threadIdx.x;
    if (i < n) p[i] = v;
}

// ---------------------------------------------------------------------------
// fp32 WMMA GEMM, compile-time K and B-transpose, RPW row tiles per wave.
// O[row][col] = act( sum_k A[row][k] * Bsel[k][col] + bias[col] )
// Bsel = BT ? B^T (B stored [col][k], row-stride ldb) : B (stored [k][col]).
// One wave; B fragments for this column tile preloaded into registers and
// reused across RPW row tiles. grid.x = ceil(rowTiles/RPW), grid.y = col tiles.
// fp32 A 16x4 fragment: lanes 0-15 hold K=k0,k0+1; lanes 16-31 hold K=k0+2,k0+3
// (row M = lane&15). fp32 B 4x16 fragment mirrored (col N = lane&15).
template <int K, bool BT, int RPW>
__global__ __launch_bounds__(32)
void wmma_gemm_f32(const float* __restrict__ A, int lda,
                   const float* __restrict__ B, int ldb,
                   float* __restrict__ O, int ldo,
                   const float* __restrict__ bias, int act, int colLimit,
                   int rowTiles) {
    constexpr int KS = K / 4;
    int l = threadIdx.x;
    int lr = l & 15;
    int half = (l < 16) ? 0 : 2;
    int colBase = blockIdx.y * 16;
    int col = colBase + lr;

    // preload the loop-invariant B fragments for this column tile
    v2f bfrag[KS];
#pragma unroll
    for (int ks = 0; ks < KS; ++ks) {
        int ka = ks * 4 + half;
        if (BT) {
            bfrag[ks].x = B[col * ldb + ka];
            bfrag[ks].y = B[col * ldb + ka + 1];
        } else {
            bfrag[ks].x = B[ka * ldb + col];
            bfrag[ks].y = B[(ka + 1) * ldb + col];
        }
    }

    float bv = (bias != nullptr && col < colLimit) ? bias[col] : 0.0f;

    for (int rt = 0; rt < RPW; ++rt) {
        int rowTile = blockIdx.x * RPW + rt;
        if (rowTile >= rowTiles) break;          // uniform branch: EXEC untouched
        int rowBase = rowTile * 16;
        int arow = rowBase + lr;

        v8f c = {};
#pragma unroll
        for (int ks = 0; ks < KS; ++ks) {
            int ka = ks * 4 + half;
            v2f a;
            a.x = A[(long)arow * lda + ka];
            a.y = A[(long)arow * lda + ka + 1];
            c = __builtin_amdgcn_wmma_f32_16x16x4_f32(false, a, false, bfrag[ks],
                                                      (short)0, c, false, false);
        }

        if (col < colLimit) {
#pragma unroll
            for (int r = 0; r < 8; ++r) {
                int row = rowBase + r + ((l < 16) ? 0 : 8);
                float v = c[r] + bv;
                if (act == 1) v = elu1(v);
                O[(long)row * ldo + col] = v;
            }
        }
    }
}

// ---------------------------------------------------------------------------
// edge scatter: agg[dst] = max(agg[dst], m[src] * ew)  (4 channels per thread)
__global__ void scatter_max_kernel(const float* __restrict__ m,
                                   const long long* __restrict__ src,
                                   const long long* __restrict__ dst,
                                   const float* __restrict__ ew,
                                   float* __restrict__ agg,
                                   int E, int C, int qshift) {
    int t = blockIdx.x * blockDim.x + threadIdx.x;
    int e = t >> qshift;
    if (e >= E) return;
    int q = (t - (e << qshift)) << 2;          // channel offset (multiple of 4)
    long long s = src[e], d = dst[e];
    float w = ew[e];
    const float4 mv = *(const float4*)(m + (long)s * C + q);
    float* base = agg + (long)d * C + q;
    atomicMaxF(base + 0, mv.x * w);
    atomicMaxF(base + 1, mv.y * w);
    atomicMaxF(base + 2, mv.z * w);
    atomicMaxF(base + 3, mv.w * w);
}

// ---------------------------------------------------------------------------
// Fused GRU cell, one wave per 16-node tile.
//   gi = sanitize(agg) @ Wih^T ; gh = h @ Whh^T    (WMMA f32 into LDS)
//   r = sig(ir+hr); z = sig(iz+hz); n = tanh(in + r*hn); h' = (1-z)*n + z*h
// act!=0 -> apply elu to h' (final propagation step of layer 2).
template <int C>
__global__ __launch_bounds__(32)
void gru_kernel(const float* __restrict__ agg,
                float* __restrict__ h,
                const float* __restrict__ Wih, const float* __restrict__ Whh,
                const float* __restrict__ bih, const float* __restrict__ bhh,
                int act) {
    constexpr int G = 3 * C;
    __shared__ float gi[16 * G];
    __shared__ float gh[16 * G];

    int l = threadIdx.x;
    int lr = l & 15;
    int half = (l < 16) ? 0 : 2;
    int rowBase = blockIdx.x * 16;
    int arow = rowBase + lr;

    for (int srcSel = 0; srcSel < 2; ++srcSel) {
        const float* A = (srcSel == 0) ? agg : h;
        const float* W = (srcSel == 0) ? Wih : Whh;
        float* out = (srcSel == 0) ? gi : gh;
#pragma unroll
        for (int ct = 0; ct < G / 16; ++ct) {
            int col = ct * 16 + lr;
            v8f c = {};
#pragma unroll
            for (int k0 = 0; k0 < C; k0 += 4) {
                int ka = k0 + half;
                float ax = A[arow * C + ka];
                float ay = A[arow * C + ka + 1];
                if (srcSel == 0) {  // segment_max left -inf for nodes with no in-edges
                    if (ax < -1.0e38f) ax = 0.0f;
                    if (ay < -1.0e38f) ay = 0.0f;
                }
                v2f a, b;
                a.x = ax; a.y = ay;
                b.x = W[col * C + ka];      // W is (3C x C), we need W^T
                b.y = W[col * C + ka + 1];
                c = __builtin_amdgcn_wmma_f32_16x16x4_f32(false, a, false, b,
                                                          (short)0, c, false, false);
            }
#pragma unroll
            for (int r = 0; r < 8; ++r) {
                int mrow = r + ((l < 16) ? 0 : 8);
                out[mrow * G + col] = c[r];
            }
        }
    }
    __syncthreads();

    for (int idx = l; idx < 16 * C; idx += 32) {
        int mrow = idx / C, j = idx % C;
        int row = rowBase + mrow;
        float ir = gi[mrow * G + j]         + bih[j];
        float iz = gi[mrow * G + C + j]     + bih[C + j];
        float in = gi[mrow * G + 2 * C + j] + bih[2 * C + j];
        float hr = gh[mrow * G + j]         + bhh[j];
        float hz = gh[mrow * G + C + j]     + bhh[C + j];
        float hn = gh[mrow * G + 2 * C + j] + bhh[2 * C + j];
        float r = 1.0f / (1.0f + expf(-(ir + hr)));
        float z = 1.0f / (1.0f + expf(-(iz + hz)));
        float n = tanhf(in + r * hn);
        float hold = h[(long)row * C + j];
        float hv = (1.0f - z) * n + z * hold;
        if (act) hv = elu1(hv);
        h[(long)row * C + j] = hv;
    }
}

// ---------------------------------------------------------------------------
// in-place log_softmax over 10 classes, one thread per node
__global__ void log_softmax10_kernel(float* __restrict__ out, int n) {
    int i = blockIdx.x * blockDim.x + threadIdx.x;
    if (i >= n) return;
    float v[10];
    float mx = -INFINITY;
#pragma unroll
    for (int c = 0; c < 10; ++c) { v[c] = out[(long)i * 10 + c]; mx = fmaxf(mx, v[c]); }
    float s = 0.0f;
#pragma unroll
    for (int c = 0; c < 10; ++c) s += expf(v[c] - mx);
    float ls = mx + logf(s);
#pragma unroll
    for (int c = 0; c < 10; ++c) out[(long)i * 10 + c] = v[c] - ls;
}

// ---------------------------------------------------------------------------
extern "C" void kernel_launch(void* const* d_in, const int* in_sizes, int n_in,
                              void* d_out, int out_size, void* d_ws, size_t ws_size,
                              hipStream_t stream) {
    const int N = N_NODES, E = N_EDGES;
    const float*     x     = (const float*)d_in[0];
    const long long* ei    = (const long long*)d_in[1];   // (2, E) int64
    const float*     ea    = (const float*)d_in[2];       // (E, 1)
    const float*     W1    = (const float*)d_in[3];       // (3,32,32)
    const float*     Wih1  = (const float*)d_in[4];       // (96,32)
    const float*     Whh1  = (const float*)d_in[5];
    const float*     bih1  = (const float*)d_in[6];
    const float*     bhh1  = (const float*)d_in[7];
    const float*     W2    = (const float*)d_in[8];       // (3,64,64)
    const float*     Wih2  = (const float*)d_in[9];       // (192,64)
    const float*     Whh2  = (const float*)d_in[10];
    const float*     bih2  = (const float*)d_in[11];
    const float*     bhh2  = (const float*)d_in[12];
    const float*     fc1w  = (const float*)d_in[13];      // (128,64)
    const float*     fc1b  = (const float*)d_in[14];
    const float*     fc2w  = (const float*)d_in[15];      // (10,128)
    const float*     fc2b  = (const float*)d_in[16];

    float* ws  = (float*)d_ws;
    float* h1  = ws;                       // N*32
    float* m   = ws + (size_t)N * 32;      // N*64
    float* agg = ws + (size_t)N * 96;      // N*64
    float* h2  = ws + (size_t)N * 160;     // N*64
    float* hid = ws;                       // N*128 (overlays h1/m/agg, all dead by fc1)

    const long long* src = ei;
    const long long* dst = ei + E;

    const int TILES = N / 16;        // 6250 exactly
    const int RPW   = 5;             // 6250 = 5 * 1250
    const int GX    = TILES / RPW;   // 1250

    // x -> h1 (pad 16 -> 32)
    pad16to32_kernel<<<(N * 32 + 255) / 256, 256, 0, stream>>>(x, h1, N);

    // ---- layer 1 (C = 32) ----
    for (int i = 0; i < 3; ++i) {
        wmma_gemm_f32<32, false, 5><<<dim3(GX, 2), 32, 0, stream>>>(
            h1, 32, W1 + (size_t)i * 32 * 32, 32, m, 32, nullptr, 0, 32, TILES);
        fill_kernel<<<((long)N * 32 + 255) / 256, 256, 0, stream>>>(agg, -INFINITY, (long)N * 32);
        scatter_max_kernel<<<((long)E * 8 + 255) / 256, 256, 0, stream>>>(m, src, dst, ea, agg, E, 32, 3);
        gru_kernel<32><<<TILES, 32, 0, stream>>>(agg, h1, Wih1, Whh1, bih1, bhh1, 0);
    }

    // elu + pad 32 -> 64
    elu_pad32to64_kernel<<<(N * 64 + 255) / 256, 256, 0, stream>>>(h1, h2, N);

    // ---- layer 2 (C = 64) ----
    for (int i = 0; i < 3; ++i) {
        wmma_gemm_f32<64, false, 5><<<dim3(GX, 4), 32, 0, stream>>>(
            h2, 64, W2 + (size_t)i * 64 * 64, 64, m, 64, nullptr, 0, 64, TILES);
        fill_kernel<<<((long)N * 64 + 255) / 256, 256, 0, stream>>>(agg, -INFINITY, (long)N * 64);
        scatter_max_kernel<<<((long)E * 16 + 255) / 256, 256, 0, stream>>>(m, src, dst, ea, agg, E, 64, 4);
        gru_kernel<64><<<TILES, 32, 0, stream>>>(agg, h2, Wih2, Whh2, bih2, bhh2, (i == 2) ? 1 : 0);
    }

    // fc1: hid = elu(h2 @ fc1w^T + fc1b)   (N x 64 -> N x 128)
    wmma_gemm_f32<64, true, 5><<<dim3(GX, 8), 32, 0, stream>>>(
        h2, 64, fc1w, 64, hid, 128, fc1b, 1, 128, TILES);
    // fc2: out = hid @ fc2w^T + fc2b       (N x 128 -> N x 10)
    wmma_gemm_f32<128, true, 5><<<dim3(GX, 1), 32, 0, stream>>>(
        hid, 128, fc2w, 128, (float*)d_out, 10, fc2b, 0, 10, TILES);
    // log_softmax in place
    log_softmax10_kernel<<<(N + 255) / 256, 256, 0, stream>>>((float*)d_out, N);
}